// DPLSTM_10033043603508
// MI455X (gfx1250) — compile-verified
//
#include <hip/hip_runtime.h>
#include <hip/hip_bf16.h>

typedef __attribute__((ext_vector_type(2))) float v2f;
typedef __attribute__((ext_vector_type(8))) float v8f;

// Problem sizes (fixed by the reference).
constexpr int Bm = 64, Sm = 512, Im = 512, Hm = 512, Lm = 2;
constexpr int G4 = 4 * Hm;                 // 2048 gate columns
// Launch shape: 64 workgroups x 256 threads = 512 wave32s (persistent kernel,
// must be co-resident for the software grid barrier).
constexpr int NBLK = 64, NTHR = 256;
constexpr int WPB = NTHR / 32;             // 8 waves per block
// Tile groups: one group = 8 consecutive 16x16 tiles sharing one A row-panel.
constexpr int T_G = (Bm / 16) * (G4 / 16); // 512 gate tiles per layer
constexpr int T_A = (Bm / 16) * (Hm / 16); // 128 adaptive-gate tiles per layer
constexpr int NG_GATE = T_G / WPB;         // 64 groups per gate segment
constexpr int NG_AG = T_A / WPB;           // 16 groups per adaptive segment
constexpr int NGRP = 2 * (NG_GATE + NG_AG);// 160 groups per phase
// LDS A-panel: 16 rows x 512 K, padded by 4 words so the 32-lane ds_load_b64
// fragment reads (row stride 516 mod 64 = 4) cover all 64 LDS banks.
constexpr int LDP = Im + 4;                // 516 words per panel row (33 KB)

// ---------------------------------------------------------------------------
// fp32 WMMA 16x16xK accumulation, A sourced from the LDS panel:
//   A layout: lanes 0-15 = row M=lane, VGPR0/1 = K,K+1; lanes 16-31 = K+2,K+3
//   B layout: lanes 0-15 = col N=lane, VGPR0/1 = rows K,K+1; lanes 16-31 = +2,+3
// ---------------------------------------------------------------------------
__device__ __forceinline__ v8f wmma_lds(const float* As /*LDS panel*/,
                                        const float* __restrict__ Bcol,
                                        int ldb, v8f acc, int l16, int half) {
  const float* ap = As + l16 * LDP + 2 * half;
  const float* bp = Bcol + (size_t)(2 * half) * ldb;
#pragma unroll 4
  for (int kk = 0; kk < Im; kk += 4) {
    v2f a;
    a.x = ap[kk];
    a.y = ap[kk + 1];
    v2f b;
    b.x = bp[(size_t)kk * ldb];
    b.y = bp[(size_t)kk * ldb + ldb];
    acc = __builtin_amdgcn_wmma_f32_16x16x4_f32(false, a, false, b,
                                                (short)0, acc, false, false);
  }
  return acc;
}

// Cooperative stage of a 16 x 512 fp32 row-panel into LDS (float4 moves).
__device__ __forceinline__ void stage_panel(float* sm,
                                            const float* __restrict__ src,
                                            size_t row_stride) {
  for (int i = threadIdx.x; i < 16 * (Im / 4); i += NTHR) {
    const int r = i >> 7;          // 128 float4 per row
    const int c4 = i & 127;
    const float4 v = *(const float4*)(src + (size_t)r * row_stride + 4 * c4);
    *(float4*)(sm + r * LDP + 4 * c4) = v;
  }
}

// C/D layout: VGPR r holds row m0 + half*8 + r, col = lane&15 (already in n).
__device__ __forceinline__ void tile_store(float* __restrict__ C, int ldc,
                                           int m0, int n, v8f acc, float bias,
                                           int half) {
  const int r0 = m0 + half * 8;
#pragma unroll
  for (int r = 0; r < 8; ++r)
    C[(size_t)(r0 + r) * ldc + n] = acc[r] + bias;
}

__device__ __forceinline__ float sigf(float v) {
  return 1.0f / (1.0f + __expf(-v));
}
__device__ __forceinline__ float tanhfast(float v) {
  float e = __expf(-2.0f * v);
  return (1.0f - e) / (1.0f + e);
}

// Sense-reversing software grid barrier (all NBLK workgroups co-resident).
__device__ __forceinline__ void grid_sync(unsigned* bar) {
  __syncthreads();
  if (threadIdx.x == 0) {
    __threadfence();
    unsigned gen = __hip_atomic_load(&bar[1], __ATOMIC_ACQUIRE,
                                     __HIP_MEMORY_SCOPE_AGENT);
    unsigned arrived = __hip_atomic_fetch_add(&bar[0], 1u, __ATOMIC_ACQ_REL,
                                              __HIP_MEMORY_SCOPE_AGENT);
    if (arrived == (unsigned)(NBLK - 1)) {
      __hip_atomic_store(&bar[0], 0u, __ATOMIC_RELAXED,
                         __HIP_MEMORY_SCOPE_AGENT);
      __hip_atomic_fetch_add(&bar[1], 1u, __ATOMIC_RELEASE,
                             __HIP_MEMORY_SCOPE_AGENT);
    } else {
      while (__hip_atomic_load(&bar[1], __ATOMIC_ACQUIRE,
                               __HIP_MEMORY_SCOPE_AGENT) == gen)
        __builtin_amdgcn_s_sleep(1);
    }
  }
  __syncthreads();
}

__global__ void dplstm_init(float* state, unsigned* bar) {
  const int tid = blockIdx.x * blockDim.x + threadIdx.x;
  const int n = 4 * Bm * Hm;  // h0, c0, h1, c1
  for (int i = tid; i < n; i += gridDim.x * blockDim.x) state[i] = 0.0f;
  if (tid < 2) bar[tid] = 0u;
}

__global__ __launch_bounds__(NTHR) void dplstm_persistent(
    const float* __restrict__ x, const float* __restrict__ W_ih,
    const float* __restrict__ b_ih, const float* __restrict__ W_hh,
    const float* __restrict__ b_hh, const float* __restrict__ W_ag,
    const float* __restrict__ b_ag, float* __restrict__ out,
    float* __restrict__ ws) {
  __shared__ float smem[16 * LDP];   // 33 KB A-panel (of 320 KB WGP LDS)

  // Workspace carve-out (~1.75 MB): gate pre-activations + state + barrier.
  float* G0 = ws;                    // [64, 2048]
  float* G1 = G0 + Bm * G4;          // [64, 2048]
  float* A0 = G1 + Bm * G4;          // [64, 512]
  float* A1 = A0 + Bm * Hm;          // [64, 512]
  float* h0 = A1 + Bm * Hm;
  float* c0 = h0 + Bm * Hm;
  float* h1 = c0 + Bm * Hm;
  float* c1 = h1 + Bm * Hm;
  unsigned* bar = (unsigned*)(c1 + Bm * Hm);

  const float* W_ih0 = W_ih;         const float* W_ih1 = W_ih + (size_t)Im * G4;
  const float* b_ih0 = b_ih;         const float* b_ih1 = b_ih + G4;
  const float* W_hh0 = W_hh;         const float* W_hh1 = W_hh + (size_t)Hm * G4;
  const float* b_hh0 = b_hh;         const float* b_hh1 = b_hh + G4;
  const float* W_ag0 = W_ag;         const float* W_ag1 = W_ag + (size_t)Hm * Hm;
  const float* b_ag0 = b_ag;         const float* b_ag1 = b_ag + Hm;

  const int lane = threadIdx.x & 31;
  const int l16 = lane & 15;
  const int half = lane >> 4;
  const int w = threadIdx.x >> 5;    // wave id in block = tile slot in group
  const int tid = blockIdx.x * blockDim.x + threadIdx.x;
  const int NTH = NBLK * NTHR;

  // Pipelined phases: phase k runs layer0 @ t=k and layer1 @ t=k-1.
  for (int k = 0; k <= Sm; ++k) {
    // -------- GEMM phase: each block owns groups of 8 tiles --------
    for (int grp = blockIdx.x; grp < NGRP; grp += NBLK) {
      if (grp < NG_GATE) {                       // layer0 gates, t=k
        if (k < Sm) {
          const int mi = grp >> 4, njg = grp & 15;
          const int n = (njg * WPB + w) * 16 + l16;
          v8f acc = {};
          // input term: x[:, k, :] @ W_ih0 (A panel from LDS)
          stage_panel(smem, x + (size_t)(mi * 16) * (Sm * Im) + (size_t)k * Im,
                      (size_t)Sm * Im);
          __syncthreads();
          acc = wmma_lds(smem, W_ih0 + n, G4, acc, l16, half);
          __syncthreads();
          // recurrent term: h0(k-1) @ W_hh0
          stage_panel(smem, h0 + (size_t)(mi * 16) * Hm, (size_t)Hm);
          __syncthreads();
          acc = wmma_lds(smem, W_hh0 + n, G4, acc, l16, half);
          __syncthreads();
          tile_store(G0, G4, mi * 16, n, acc, b_ih0[n] + b_hh0[n], half);
        }
      } else if (grp < NG_GATE + NG_AG) {        // layer0 adaptive gate, t=k
        if (k < Sm) {
          const int ga = grp - NG_GATE;
          const int mi = ga >> 2, njg = ga & 3;
          const int n = (njg * WPB + w) * 16 + l16;
          v8f acc = {};
          stage_panel(smem, h0 + (size_t)(mi * 16) * Hm, (size_t)Hm);
          __syncthreads();
          acc = wmma_lds(smem, W_ag0 + n, Hm, acc, l16, half);
          __syncthreads();
          tile_store(A0, Hm, mi * 16, n, acc, b_ag0[n], half);
        }
      } else if (grp < 2 * NG_GATE + NG_AG) {    // layer1 gates, t=k-1
        if (k >= 1) {
          const int gg = grp - (NG_GATE + NG_AG);
          const int mi = gg >> 4, njg = gg & 15;
          const int n = (njg * WPB + w) * 16 + l16;
          v8f acc = {};
          // input term: h0(k-1) (= layer0 output @ t=k-1) @ W_ih1
          stage_panel(smem, h0 + (size_t)(mi * 16) * Hm, (size_t)Hm);
          __syncthreads();
          acc = wmma_lds(smem, W_ih1 + n, G4, acc, l16, half);
          __syncthreads();
          // recurrent term: h1(k-2) @ W_hh1
          stage_panel(smem, h1 + (size_t)(mi * 16) * Hm, (size_t)Hm);
          __syncthreads();
          acc = wmma_lds(smem, W_hh1 + n, G4, acc, l16, half);
          __syncthreads();
          tile_store(G1, G4, mi * 16, n, acc, b_ih1[n] + b_hh1[n], half);
        }
      } else {                                   // layer1 adaptive gate, t=k-1
        if (k >= 1) {
          const int ga = grp - (2 * NG_GATE + NG_AG);
          const int mi = ga >> 2, njg = ga & 3;
          const int n = (njg * WPB + w) * 16 + l16;
          v8f acc = {};
          stage_panel(smem, h1 + (size_t)(mi * 16) * Hm, (size_t)Hm);
          __syncthreads();
          acc = wmma_lds(smem, W_ag1 + n, Hm, acc, l16, half);
          __syncthreads();
          tile_store(A1, Hm, mi * 16, n, acc, b_ag1[n], half);
        }
      }
    }
    grid_sync(bar);

    // -------- pointwise cell update phase --------
    if (k < Sm) {                              // layer0 -> h0(k), c0(k)
      for (int e = tid; e < Bm * Hm; e += NTH) {
        const int b = e >> 9, hh = e & (Hm - 1);
        const float* g = G0 + (size_t)b * G4;
        const float iv = sigf(g[hh]);
        const float fv = sigf(g[Hm + hh]);
        const float gv = tanhfast(g[2 * Hm + hh]);
        const float ov = sigf(g[3 * Hm + hh]);
        const float av = sigf(A0[e]);
        const float cn = av * (fv * c0[e]) + (1.0f - av) * (iv * gv);
        c0[e] = cn;
        h0[e] = ov * tanhfast(cn);
      }
    }
    if (k >= 1) {                              // layer1 -> h1(k-1), out[:,k-1,:]
      for (int e = tid; e < Bm * Hm; e += NTH) {
        const int b = e >> 9, hh = e & (Hm - 1);
        const float* g = G1 + (size_t)b * G4;
        const float iv = sigf(g[hh]);
        const float fv = sigf(g[Hm + hh]);
        const float gv = tanhfast(g[2 * Hm + hh]);
        const float ov = sigf(g[3 * Hm + hh]);
        const float av = sigf(A1[e]);
        const float cn = av * (fv * c1[e]) + (1.0f - av) * (iv * gv);
        const float hn = ov * tanhfast(cn);
        c1[e] = cn;
        h1[e] = hn;
        out[((size_t)b * Sm + (k - 1)) * Hm + hh] = hn;
      }
    }
    grid_sync(bar);
  }

  // Final carries: h_n [1,L,B,H] then c_n [1,L,B,H] appended after out.
  const size_t OUT0 = (size_t)Bm * Sm * Hm;
  for (int i = tid; i < Lm * Bm * Hm; i += NTH) {
    const int l = i / (Bm * Hm);
    const int r = i - l * (Bm * Hm);
    out[OUT0 + i] = (l == 0 ? h0 : h1)[r];
    out[OUT0 + (size_t)Lm * Bm * Hm + i] = (l == 0 ? c0 : c1)[r];
  }
}

extern "C" void kernel_launch(void* const* d_in, const int* in_sizes, int n_in,
                              void* d_out, int out_size, void* d_ws,
                              size_t ws_size, hipStream_t stream) {
  const float* x    = (const float*)d_in[0];
  const float* W_ih = (const float*)d_in[1];
  const float* b_ih = (const float*)d_in[2];
  const float* W_hh = (const float*)d_in[3];
  const float* b_hh = (const float*)d_in[4];
  const float* W_ag = (const float*)d_in[5];
  const float* b_ag = (const float*)d_in[6];
  float* out = (float*)d_out;
  float* ws  = (float*)d_ws;

  // State region (h0,c0,h1,c1) and barrier words inside the workspace.
  float* state  = ws + 2 * (size_t)Bm * G4 + 2 * (size_t)Bm * Hm;
  unsigned* bar = (unsigned*)(state + 4 * (size_t)Bm * Hm);

  hipLaunchKernelGGL(dplstm_init, dim3(64), dim3(256), 0, stream, state, bar);
  hipLaunchKernelGGL(dplstm_persistent, dim3(NBLK), dim3(NTHR), 0, stream, x,
                     W_ih, b_ih, W_hh, b_hh, W_ag, b_ag, out, ws);
}